// Attention_2534030704919
// MI455X (gfx1250) — compile-verified
//
#include <hip/hip_runtime.h>

// Problem constants (match reference)
#define B_DIM 32
#define N_DIM 1024
#define D_DIM 256   // DIN == DOUT

typedef __attribute__((ext_vector_type(2))) float v2f;
typedef __attribute__((ext_vector_type(8))) float v8f;

// V_WMMA_F32_16X16X4_F32: D(16x16 f32) = A(16x4 f32) * B(4x16 f32) + C
__device__ __forceinline__ v8f wmma_f32_16x16x4(v2f a, v2f b, v8f c) {
    return __builtin_amdgcn_wmma_f32_16x16x4_f32(
        false, a, false, b, (short)0, c, false, false);
}

// ---------------------------------------------------------------------------
// Kernel 1: q/k/v = x @ W^T + bias.
// One wave -> one 64x64 macro-tile (4x4 grid of 16x16 WMMA tiles) of one of
// q/k/v.  Per K-step: 4 A-frags + 4 B-frags (2KB) -> 16 WMMAs (32K FLOP).
// ---------------------------------------------------------------------------
__global__ __launch_bounds__(256) void qkv_proj_kernel(
    const float* __restrict__ x,
    const float* __restrict__ Wq, const float* __restrict__ bq,
    const float* __restrict__ Wk, const float* __restrict__ bk,
    const float* __restrict__ Wv, const float* __restrict__ bv,
    float* __restrict__ q, float* __restrict__ k, float* __restrict__ v)
{
    const int lane = threadIdx.x & 31;
    const int gw   = blockIdx.x * 8 + (threadIdx.x >> 5);

    const int colMacros    = (D_DIM / 16) / 4;             // 4
    const int rowMacros    = ((B_DIM * N_DIM) / 16) / 4;   // 512
    const int tilesPerMat  = rowMacros * colMacros;        // 2048

    const int mat  = gw / tilesPerMat;                     // 0=q 1=k 2=v
    const int t    = gw % tilesPerMat;
    const int rowM = t / colMacros;
    const int colM = t % colMacros;

    const float* W    = (mat == 0) ? Wq : (mat == 1) ? Wk : Wv;
    const float* bias = (mat == 0) ? bq : (mat == 1) ? bk : bv;
    float*       out  = (mat == 0) ? q  : (mat == 1) ? k  : v;

    const int l15  = lane & 15;
    const int hi   = lane >> 4;        // 0: K={0,1}, 1: K={2,3}
    const int row0 = rowM * 64;
    const int col0 = colM * 64;

    const float* arow[4];
    const float* brow[4];
#pragma unroll
    for (int i = 0; i < 4; ++i) {
        arow[i] = x + (size_t)(row0 + 16 * i + l15) * D_DIM + 2 * hi;
        brow[i] = W + (size_t)(col0 + 16 * i + l15) * D_DIM + 2 * hi;
    }

    v8f acc[4][4];
#pragma unroll
    for (int i = 0; i < 4; ++i)
#pragma unroll
        for (int j = 0; j < 4; ++j)
            acc[i][j] = (v8f){};

    for (int kk = 0; kk < D_DIM; kk += 4) {
        v2f a[4], b[4];
#pragma unroll
        for (int i = 0; i < 4; ++i) {
            a[i] = (v2f){ arow[i][kk], arow[i][kk + 1] };
            b[i] = (v2f){ brow[i][kk], brow[i][kk + 1] };
        }
#pragma unroll
        for (int i = 0; i < 4; ++i)
#pragma unroll
            for (int j = 0; j < 4; ++j)
                acc[i][j] = wmma_f32_16x16x4(a[i], b[j], acc[i][j]);
    }

    float bb[4];
#pragma unroll
    for (int j = 0; j < 4; ++j) bb[j] = bias[col0 + 16 * j + l15];

#pragma unroll
    for (int i = 0; i < 4; ++i)
#pragma unroll
        for (int j = 0; j < 4; ++j)
#pragma unroll
            for (int r = 0; r < 8; ++r) {
                const int rr = r + 8 * hi;   // C/D: VGPR r -> rows r / r+8
                out[(size_t)(row0 + 16 * i + rr) * D_DIM + col0 + 16 * j + l15] =
                    acc[i][j][r] + bb[j];
            }
}

// ---------------------------------------------------------------------------
// Kernel 2: logits[b,n,m] = sum_d q[b,n,d] * k[b,m,d].
// One wave -> 64x64 macro-tile of logits for one batch (4x4 blocking).
// ---------------------------------------------------------------------------
__global__ __launch_bounds__(256) void logits_kernel(
    const float* __restrict__ q, const float* __restrict__ k,
    float* __restrict__ logits)
{
    const int lane = threadIdx.x & 31;
    const int gw   = blockIdx.x * 8 + (threadIdx.x >> 5);

    const int mMacros = (N_DIM / 16) / 4;   // 16
    const int nMacros = (N_DIM / 16) / 4;   // 16
    const int b  = gw / (nMacros * mMacros);
    const int t  = gw % (nMacros * mMacros);
    const int nM = t / mMacros;
    const int mM = t % mMacros;

    const int l15 = lane & 15;
    const int hi  = lane >> 4;
    const int n0  = nM * 64;
    const int m0  = mM * 64;

    const float* qrow[4];
    const float* krow[4];
#pragma unroll
    for (int i = 0; i < 4; ++i) {
        qrow[i] = q + ((size_t)b * N_DIM + n0 + 16 * i + l15) * D_DIM + 2 * hi;
        krow[i] = k + ((size_t)b * N_DIM + m0 + 16 * i + l15) * D_DIM + 2 * hi;
    }

    v8f acc[4][4];
#pragma unroll
    for (int i = 0; i < 4; ++i)
#pragma unroll
        for (int j = 0; j < 4; ++j)
            acc[i][j] = (v8f){};

    for (int kk = 0; kk < D_DIM; kk += 4) {
        v2f a[4], bfr[4];
#pragma unroll
        for (int i = 0; i < 4; ++i) {
            a[i]   = (v2f){ qrow[i][kk], qrow[i][kk + 1] };
            bfr[i] = (v2f){ krow[i][kk], krow[i][kk + 1] };
        }
#pragma unroll
        for (int i = 0; i < 4; ++i)
#pragma unroll
            for (int j = 0; j < 4; ++j)
                acc[i][j] = wmma_f32_16x16x4(a[i], bfr[j], acc[i][j]);
    }

    float* out = logits + (size_t)b * N_DIM * N_DIM;
#pragma unroll
    for (int i = 0; i < 4; ++i)
#pragma unroll
        for (int j = 0; j < 4; ++j)
#pragma unroll
            for (int r = 0; r < 8; ++r) {
                const int rr = r + 8 * hi;
                out[(size_t)(n0 + 16 * i + rr) * N_DIM + m0 + 16 * j + l15] =
                    acc[i][j][r];
            }
}

// ---------------------------------------------------------------------------
// Kernel 3: cross-BATCH softmax statistics.  For each (n,m):
//   maxl[n,m]   = max_b logits[b,n,m]
//   rdenom[n,m] = 1 / sum_b exp(logits[b,n,m] - maxl[n,m])
// logits is L2-resident (128 MB vs 192 MB L2), so the double pass is cheap.
// ---------------------------------------------------------------------------
__global__ __launch_bounds__(256) void softmax_stats_kernel(
    const float* __restrict__ logits,
    float* __restrict__ maxl, float* __restrict__ rdenom)
{
    const size_t idx    = (size_t)blockIdx.x * blockDim.x + threadIdx.x; // n*N+m
    const size_t stride = (size_t)N_DIM * N_DIM;

    float mx = -3.0e38f;
#pragma unroll
    for (int b = 0; b < B_DIM; ++b)
        mx = fmaxf(mx, logits[(size_t)b * stride + idx]);

    float s = 0.0f;
#pragma unroll
    for (int b = 0; b < B_DIM; ++b)
        s += __expf(logits[(size_t)b * stride + idx] - mx);

    maxl[idx]   = mx;
    rdenom[idx] = 1.0f / s;
}

// ---------------------------------------------------------------------------
// Kernel 4: out[b,n,:] = sum_m softmax(l)[b,n,m] * v[b,m,:].
// One wave -> 64x64 macro-tile of out (4 n-tiles x 4 d-tiles).  The attn
// A-fragments are built in-register (exp(l - max) * rdenom), so the 128 MB
// attention matrix is never materialized; exps co-issue alongside WMMAs.
// ---------------------------------------------------------------------------
__global__ __launch_bounds__(256) void attnv_kernel(
    const float* __restrict__ logits, const float* __restrict__ maxl,
    const float* __restrict__ rdenom, const float* __restrict__ v,
    float* __restrict__ out)
{
    const int lane = threadIdx.x & 31;
    const int gw   = blockIdx.x * 8 + (threadIdx.x >> 5);

    const int dMacros = (D_DIM / 16) / 4;   // 4
    const int nMacros = (N_DIM / 16) / 4;   // 16
    const int b  = gw / (nMacros * dMacros);
    const int t  = gw % (nMacros * dMacros);
    const int nM = t / dMacros;
    const int dM = t % dMacros;

    const int l15 = lane & 15;
    const int hi  = lane >> 4;
    const int n0  = nM * 64;
    const int d0  = dM * 64;

    const float* lrow[4];
    const float* mrow[4];
    const float* drow[4];
#pragma unroll
    for (int i = 0; i < 4; ++i) {
        const int n = n0 + 16 * i + l15;
        lrow[i] = logits + (size_t)b * N_DIM * N_DIM + (size_t)n * N_DIM + 2 * hi;
        mrow[i] = maxl   + (size_t)n * N_DIM + 2 * hi;
        drow[i] = rdenom + (size_t)n * N_DIM + 2 * hi;
    }

    v8f acc[4][4];
#pragma unroll
    for (int i = 0; i < 4; ++i)
#pragma unroll
        for (int j = 0; j < 4; ++j)
            acc[i][j] = (v8f){};

    for (int m = 0; m < N_DIM; m += 4) {
        v2f a[4];
#pragma unroll
        for (int i = 0; i < 4; ++i) {
            a[i].x = __expf(lrow[i][m]     - mrow[i][m])     * drow[i][m];
            a[i].y = __expf(lrow[i][m + 1] - mrow[i][m + 1]) * drow[i][m + 1];
        }
        v2f bfr[4];
#pragma unroll
        for (int j = 0; j < 4; ++j) {
            // B[K][l15] = v[b][m + 2*hi + K'][d0 + 16*j + l15]
            const float* vb =
                v + ((size_t)b * N_DIM + m + 2 * hi) * D_DIM + d0 + 16 * j + l15;
            bfr[j] = (v2f){ vb[0], vb[D_DIM] };
        }
#pragma unroll
        for (int i = 0; i < 4; ++i)
#pragma unroll
            for (int j = 0; j < 4; ++j)
                acc[i][j] = wmma_f32_16x16x4(a[i], bfr[j], acc[i][j]);
    }

#pragma unroll
    for (int i = 0; i < 4; ++i)
#pragma unroll
        for (int j = 0; j < 4; ++j)
#pragma unroll
            for (int r = 0; r < 8; ++r) {
                const int rr = r + 8 * hi;
                out[((size_t)b * N_DIM + n0 + 16 * i + rr) * D_DIM + d0 + 16 * j + l15] =
                    acc[i][j][r];
            }
}

// ---------------------------------------------------------------------------
extern "C" void kernel_launch(void* const* d_in, const int* in_sizes, int n_in,
                              void* d_out, int out_size, void* d_ws, size_t ws_size,
                              hipStream_t stream) {
    const float* x  = (const float*)d_in[0];
    const float* Wq = (const float*)d_in[1];
    const float* bq = (const float*)d_in[2];
    const float* Wk = (const float*)d_in[3];
    const float* bk = (const float*)d_in[4];
    const float* Wv = (const float*)d_in[5];
    const float* bv = (const float*)d_in[6];
    float* out = (float*)d_out;

    // Workspace layout (floats): q | k | v | logits | maxl | rdenom
    const size_t qkvElems   = (size_t)B_DIM * N_DIM * D_DIM;   // 8M
    const size_t logitElems = (size_t)B_DIM * N_DIM * N_DIM;   // 32M
    float* q      = (float*)d_ws;
    float* k      = q + qkvElems;
    float* v      = k + qkvElems;
    float* logits = v + qkvElems;
    float* maxl   = logits + logitElems;
    float* rden   = maxl + (size_t)N_DIM * N_DIM;

    // 1) QKV projection: 3 * 512 * 4 macro-tiles, 8 waves/block
    {
        const int waves = 3 * (((B_DIM * N_DIM) / 16) / 4) * ((D_DIM / 16) / 4);
        qkv_proj_kernel<<<waves / 8, 256, 0, stream>>>(x, Wq, bq, Wk, bk, Wv, bv,
                                                       q, k, v);
    }
    // 2) logits: 32 * 16 * 16 macro-tiles
    {
        const int waves = B_DIM * ((N_DIM / 16) / 4) * ((N_DIM / 16) / 4);
        logits_kernel<<<waves / 8, 256, 0, stream>>>(q, k, logits);
    }
    // 3) cross-batch softmax stats: one thread per (n,m)
    {
        const int threads = N_DIM * N_DIM;
        softmax_stats_kernel<<<threads / 256, 256, 0, stream>>>(logits, maxl, rden);
    }
    // 4) attn @ v with in-register softmax: 32 * 16 * 4 macro-tiles
    {
        const int waves = B_DIM * ((N_DIM / 16) / 4) * ((D_DIM / 16) / 4);
        attnv_kernel<<<waves / 8, 256, 0, stream>>>(logits, maxl, rden, v, out);
    }
}